// GCN_11665131176207
// MI455X (gfx1250) — compile-verified
//
#include <hip/hip_runtime.h>
#include <math.h>

typedef float v2f __attribute__((ext_vector_type(2)));
typedef float v8f __attribute__((ext_vector_type(8)));

#define F_IN 128

// ---------------- degree / norm ----------------

__global__ void k_init_deg(float* __restrict__ deg, int n) {
  int i = blockIdx.x * blockDim.x + threadIdx.x;
  if (i < n) deg[i] = 1.0f;  // self-loop
}

__global__ void k_count_deg(const long long* __restrict__ dst, float* __restrict__ deg, int e) {
  int i = blockIdx.x * blockDim.x + threadIdx.x;
  if (i < e) atomicAdd(&deg[(int)dst[i]], 1.0f);
}

__global__ void k_dinv(const float* __restrict__ deg, float* __restrict__ dinv, int n) {
  int i = blockIdx.x * blockDim.x + threadIdx.x;
  if (i < n) {
    float d = deg[i];
    dinv[i] = (d > 0.0f) ? rsqrtf(fmaxf(d, 1e-12f)) : 0.0f;
  }
}

// ---------------- W2 zero-pad to 16x16 ----------------

__global__ void k_pad_w2(const float* __restrict__ W2, float* __restrict__ Wp) {
  int i = threadIdx.x;           // 256 threads, 1 block
  int r = i >> 4, c = i & 15;
  Wp[i] = (c < 12) ? W2[r * 12 + c] : 0.0f;
}

// ---------------- GEMM1: H1[N,16] = X[N,128] @ W1[128,16] via V_WMMA_F32_16X16X4_F32 ----------------

__global__ void k_gemm1(const float* __restrict__ X, const float* __restrict__ W1,
                        float* __restrict__ H1, int n_tiles, int n) {
  int lane = threadIdx.x & 31;
  int wave = threadIdx.x >> 5;
  int tile = blockIdx.x * (blockDim.x >> 5) + wave;
  if (tile >= n_tiles) return;                 // wave-uniform: EXEC stays all-ones
  int half = lane >> 4;                        // 0 or 1 (K half)
  int l16  = lane & 15;
  int rowA = tile * 16 + l16;
  if (rowA >= n) rowA = n - 1;                 // clamp loads (stores guarded below)
  const float* xrow = X + (size_t)rowA * F_IN;

  v8f c = {0.f, 0.f, 0.f, 0.f, 0.f, 0.f, 0.f, 0.f};
#pragma unroll 4
  for (int k = 0; k < F_IN / 4; ++k) {
    int kk = k * 4 + half * 2;                 // even -> 8B aligned
    v2f a = *(const v2f*)(xrow + kk);
    v2f b;
    b.x = W1[kk * 16 + l16];
    b.y = W1[(kk + 1) * 16 + l16];
    c = __builtin_amdgcn_wmma_f32_16x16x4_f32(false, a, false, b, (short)0, c, false, false);
  }
  // D layout: VGPR j -> row M=j+8*half, col=l16
  if (tile * 16 + 16 <= n) {                   // full tile: branchless stores
    float* out = H1 + (size_t)(tile * 16 + 8 * half) * 16 + l16;
#pragma unroll
    for (int j = 0; j < 8; ++j) out[j * 16] = c[j];
  } else {
#pragma unroll
    for (int j = 0; j < 8; ++j) {
      int m = tile * 16 + j + 8 * half;
      if (m < n) H1[(size_t)m * 16 + l16] = c[j];
    }
  }
}

// ---------------- GEMM2: H2p[N,16] = Z1[N,16] @ Wp[16,16] (cols 12..15 zero) ----------------

__global__ void k_gemm2(const float* __restrict__ Z1, const float* __restrict__ Wp,
                        float* __restrict__ H2p, int n_tiles, int n) {
  int lane = threadIdx.x & 31;
  int wave = threadIdx.x >> 5;
  int tile = blockIdx.x * (blockDim.x >> 5) + wave;
  if (tile >= n_tiles) return;
  int half = lane >> 4;
  int l16  = lane & 15;
  int rowA = tile * 16 + l16;
  if (rowA >= n) rowA = n - 1;
  const float* zrow = Z1 + (size_t)rowA * 16;

  v8f c = {0.f, 0.f, 0.f, 0.f, 0.f, 0.f, 0.f, 0.f};
#pragma unroll
  for (int k = 0; k < 4; ++k) {
    int kk = k * 4 + half * 2;
    v2f a = *(const v2f*)(zrow + kk);
    v2f b;
    b.x = Wp[kk * 16 + l16];
    b.y = Wp[(kk + 1) * 16 + l16];
    c = __builtin_amdgcn_wmma_f32_16x16x4_f32(false, a, false, b, (short)0, c, false, false);
  }
  if (tile * 16 + 16 <= n) {
    float* out = H2p + (size_t)(tile * 16 + 8 * half) * 16 + l16;
#pragma unroll
    for (int j = 0; j < 8; ++j) out[j * 16] = c[j];
  } else {
#pragma unroll
    for (int j = 0; j < 8; ++j) {
      int m = tile * 16 + j + 8 * half;
      if (m < n) H2p[(size_t)m * 16 + l16] = c[j];
    }
  }
}

// ---------------- self-loop init of accumulators (stride-16 rows, float4) ----------------

__global__ void k_selfloop16(const float* __restrict__ H, const float* __restrict__ dinv,
                             float* __restrict__ acc, int n) {
  int t = blockIdx.x * blockDim.x + threadIdx.x;   // n*4 threads, 4 cols each
  if (t >= n * 4) return;
  int i = t >> 2, q = t & 3;
  float w = dinv[i]; w *= w;
  float4 v = *((const float4*)(H + (size_t)i * 16) + q);
  float4 r; r.x = v.x * w; r.y = v.y * w; r.z = v.z * w; r.w = v.w * w;
  *((float4*)(acc + (size_t)i * 16) + q) = r;
}

// ---------------- edge scatter (the bandwidth-bound core) ----------------

__global__ void k_scatter16(const long long* __restrict__ src, const long long* __restrict__ dst,
                            const float* __restrict__ dinv, const float* __restrict__ H,
                            float* __restrict__ acc, int e) {
  int t = blockIdx.x * blockDim.x + threadIdx.x;   // e*4 threads, 4 cols each
  if (t >= e * 4) return;
  int ei = t >> 2, q = t & 3;
  int s = (int)src[ei], d = (int)dst[ei];
  float norm = dinv[s] * dinv[d];
  float4 v = *((const float4*)(H + (size_t)s * 16) + q);   // global_load_b128 gather
  float* a = acc + (size_t)d * 16 + q * 4;
  atomicAdd(a + 0, v.x * norm);
  atomicAdd(a + 1, v.y * norm);
  atomicAdd(a + 2, v.z * norm);
  atomicAdd(a + 3, v.w * norm);
}

// 3 threads/edge, each a float4 chunk of the 12 live columns (stride-16 padded rows)
__global__ void k_scatter12p(const long long* __restrict__ src, const long long* __restrict__ dst,
                             const float* __restrict__ dinv, const float* __restrict__ H,
                             float* __restrict__ acc, int e) {
  int t = blockIdx.x * blockDim.x + threadIdx.x;   // e*3 threads
  if (t >= e * 3) return;
  int ei = t / 3, q = t - ei * 3;                  // q in 0..2 -> cols 4q..4q+3
  int s = (int)src[ei], d = (int)dst[ei];
  float norm = dinv[s] * dinv[d];
  float4 v = *((const float4*)(H + (size_t)s * 16) + q);
  float* a = acc + (size_t)d * 16 + q * 4;
  atomicAdd(a + 0, v.x * norm);
  atomicAdd(a + 1, v.y * norm);
  atomicAdd(a + 2, v.z * norm);
  atomicAdd(a + 3, v.w * norm);
}

// ---------------- epilogues ----------------

__global__ void k_bias_relu(float* __restrict__ acc, const float* __restrict__ b, int n) {
  int t = blockIdx.x * blockDim.x + threadIdx.x;   // n*16
  if (t >= n * 16) return;
  float v = acc[t] + b[t & 15];
  acc[t] = v > 0.0f ? v : 0.0f;
}

__global__ void k_logsoftmax(const float* __restrict__ accp, const float* __restrict__ b2,
                             float* __restrict__ out, int n) {
  int i = blockIdx.x * blockDim.x + threadIdx.x;
  if (i >= n) return;
  float v[12];
  float m = -INFINITY;
#pragma unroll
  for (int c = 0; c < 12; ++c) {
    v[c] = accp[(size_t)i * 16 + c] + b2[c];       // padded rows, read 12 live cols
    m = fmaxf(m, v[c]);
  }
  float s = 0.0f;
#pragma unroll
  for (int c = 0; c < 12; ++c) s += expf(v[c] - m);
  float lse = m + logf(s);
#pragma unroll
  for (int c = 0; c < 12; ++c) out[(size_t)i * 12 + c] = v[c] - lse;
}

// ---------------- launcher ----------------

extern "C" void kernel_launch(void* const* d_in, const int* in_sizes, int n_in,
                              void* d_out, int out_size, void* d_ws, size_t ws_size,
                              hipStream_t stream) {
  const float*     x   = (const float*)d_in[0];
  const long long* ei  = (const long long*)d_in[1];
  const float*     W1  = (const float*)d_in[2];
  const float*     b1  = (const float*)d_in[3];
  const float*     W2  = (const float*)d_in[4];
  const float*     b2  = (const float*)d_in[5];
  float*           out = (float*)d_out;

  const int N = in_sizes[0] / F_IN;
  const int E = in_sizes[1] / 2;
  const long long* src = ei;
  const long long* dst = ei + E;

  // carve workspace
  char* p = (char*)d_ws;
  auto alloc = [&](size_t bytes) -> float* {
    char* r = p;
    p += (bytes + 255) & ~(size_t)255;
    return (float*)r;
  };
  float* deg  = alloc((size_t)N * 4);
  float* dinv = alloc((size_t)N * 4);
  float* H1   = alloc((size_t)N * 16 * 4);
  float* Z1   = alloc((size_t)N * 16 * 4);   // acc1, becomes relu output in place
  float* H2p  = alloc((size_t)N * 16 * 4);   // layer-2 pre-activation, padded width 16
  float* A2p  = alloc((size_t)N * 16 * 4);   // layer-2 accumulator, padded width 16
  float* Wp   = alloc(16 * 16 * 4);

  const int B = 256;                 // 8 waves / block (wave32)
  const int wpb = B / 32;
  const int tiles = (N + 15) / 16;

  k_init_deg <<<(N + B - 1) / B, B, 0, stream>>>(deg, N);
  k_count_deg<<<(E + B - 1) / B, B, 0, stream>>>(dst, deg, E);
  k_dinv     <<<(N + B - 1) / B, B, 0, stream>>>(deg, dinv, N);
  k_pad_w2   <<<1, 256, 0, stream>>>(W2, Wp);

  // layer 1
  k_gemm1     <<<(tiles + wpb - 1) / wpb, B, 0, stream>>>(x, W1, H1, tiles, N);
  k_selfloop16<<<(N * 4 + B - 1) / B, B, 0, stream>>>(H1, dinv, Z1, N);
  k_scatter16 <<<(E * 4 + B - 1) / B, B, 0, stream>>>(src, dst, dinv, H1, Z1, E);
  k_bias_relu <<<(N * 16 + B - 1) / B, B, 0, stream>>>(Z1, b1, N);

  // layer 2 (padded stride-16 buffers; cols 12..15 are zero throughout)
  k_gemm2     <<<(tiles + wpb - 1) / wpb, B, 0, stream>>>(Z1, Wp, H2p, tiles, N);
  k_selfloop16<<<(N * 4 + B - 1) / B, B, 0, stream>>>(H2p, dinv, A2p, N);
  k_scatter12p<<<(E * 3 + B - 1) / B, B, 0, stream>>>(src, dst, dinv, H2p, A2p, E);
  k_logsoftmax<<<(N + B - 1) / B, B, 0, stream>>>(A2p, b2, out, N);
}